// KeplerQuantizer_24781961298393
// MI455X (gfx1250) — compile-verified
//
#include <hip/hip_runtime.h>
#include <hip/hip_bf16.h>
#include <stddef.h>

typedef __attribute__((ext_vector_type(2))) float v2f;
typedef __attribute__((ext_vector_type(8))) float v8f;

#define B_   8
#define C_   256
#define H_   32
#define W_   32
#define P_   4
#define ED   64            // e_dim
#define NE   8192          // codebook entries
#define NROW 32768         // B*P*H*W flattened zf rows
#define NOUT 2097152       // B*C*H*W

#define WAVES   8
#define BLOCK   256
#define MTILE   128        // rows per block (8 waves * 16)
#define NCHUNK  128        // codebook entries staged in LDS per iteration
#define NITER   (NE / NCHUNK)
#define LDSTR   66         // padded row stride (floats) to avoid bank conflicts
#define STAGE4  ((NCHUNK * ED / 4) / BLOCK)   // float4 per thread per chunk (8)

// ---------------------------------------------------------------------------
// Kernel 1: per-entry codebook squared norms -> ws
// ---------------------------------------------------------------------------
__global__ __launch_bounds__(BLOCK)
void vq_norms_kernel(const float* __restrict__ cb, float* __restrict__ norms) {
    int e = blockIdx.x * BLOCK + threadIdx.x;       // 0..NE-1
    const float4* c4 = (const float4*)(cb + (size_t)e * ED);
    float s = 0.f;
    #pragma unroll
    for (int i = 0; i < ED / 4; ++i) {
        float4 v = c4[i];
        s += v.x * v.x + v.y * v.y + v.z * v.z + v.w * v.w;
    }
    norms[e] = s;
}

// ---------------------------------------------------------------------------
// Kernel 2: fused distance-GEMM (FP32 WMMA 16x16x4) + argmin over 8192 entries
// Software-pipelined: next chunk is prefetched into VGPRs while the current
// LDS-resident chunk feeds the WMMA pipeline.
// ---------------------------------------------------------------------------
__global__ __launch_bounds__(BLOCK)
void vq_argmin_kernel(const float* __restrict__ z,
                      const float* __restrict__ cb,
                      const float* __restrict__ norms,
                      int* __restrict__ out_idx) {
    __shared__ float tile[NCHUNK * LDSTR];   // 33792 B codebook stage
    __shared__ float ntile[NCHUNK];          // staged norms

    const int tid  = threadIdx.x;
    const int lane = tid & 31;
    const int wave = tid >> 5;
    const int half = lane >> 4;              // K-pair selector (ISA A/B layout)
    const int lrow = lane & 15;              // M (for A) / N (for B) sub-index
    const int rowbase = blockIdx.x * MTILE + wave * 16;

    // --- Load resident A fragments: zf rows [rowbase, rowbase+16), K=0..63.
    // zf[j,d] = z[b, p*64+d, h, w],  j = ((b*4+p)*32+h)*32+w
    const int j  = rowbase + lrow;
    const int ww = j & 31, hh = (j >> 5) & 31, pp = (j >> 10) & 3, bb = j >> 12;
    const float* zrow = z + ((((size_t)bb * C_ + pp * ED) * H_ + hh) * W_ + ww);
    // element d lives at zrow + d*H_*W_
    v2f afrag[16];
    #pragma unroll
    for (int kk = 0; kk < 16; ++kk) {
        const int k0 = kk * 4 + half * 2;
        afrag[kk].x = zrow[(size_t)k0 * (H_ * W_)];
        afrag[kk].y = zrow[(size_t)(k0 + 1) * (H_ * W_)];
    }

    float best[8];
    int   bidx[8];
    #pragma unroll
    for (int g = 0; g < 8; ++g) { best[g] = 3.4e38f; bidx[g] = 0; }

    // --- Pipeline prologue: fetch chunk 0 into staging registers.
    float4 creg[STAGE4];
    float  nreg = 0.f;
    {
        const float4* src = (const float4*)cb;
        #pragma unroll
        for (int i = 0; i < STAGE4; ++i) creg[i] = src[tid + i * BLOCK];
        if (tid < NCHUNK) nreg = norms[tid];
    }

    for (int c = 0; c < NITER; ++c) {
        const int n0 = c * NCHUNK;

        __syncthreads();   // previous chunk's compute finished reading LDS
        // Drain staging registers into padded LDS rows.
        #pragma unroll
        for (int i = 0; i < STAGE4; ++i) {
            const int t    = tid + i * BLOCK;     // float4 index 0..2047
            const int ent  = t >> 4;              // 16 float4 per entry
            const int col4 = t & 15;
            float* d = &tile[ent * LDSTR + col4 * 4];
            d[0] = creg[i].x; d[1] = creg[i].y; d[2] = creg[i].z; d[3] = creg[i].w;
        }
        if (tid < NCHUNK) ntile[tid] = nreg;
        __syncthreads();

        // Prefetch next chunk while the WMMA pipeline below hides the latency.
        if (c + 1 < NITER) {
            const float4* src = (const float4*)(cb + (size_t)(c + 1) * NCHUNK * ED);
            #pragma unroll
            for (int i = 0; i < STAGE4; ++i) creg[i] = src[tid + i * BLOCK];
            if (tid < NCHUNK) nreg = norms[(c + 1) * NCHUNK + tid];
        }

        #pragma unroll
        for (int sub = 0; sub < NCHUNK / 16; ++sub) {
            // B fragment: B[k][n] = cb[n0+sub*16+n][k]; mirrors A layout in N.
            const float* bbase = &tile[(sub * 16 + lrow) * LDSTR + half * 2];
            v8f acc = {0.f, 0.f, 0.f, 0.f, 0.f, 0.f, 0.f, 0.f};
            #pragma unroll
            for (int kk = 0; kk < 16; ++kk) {
                const v2f bfrag = *(const v2f*)(bbase + kk * 4);
                acc = __builtin_amdgcn_wmma_f32_16x16x4_f32(
                        false, afrag[kk], false, bfrag,
                        (short)0, acc, false, false);
            }
            const int   n  = n0 + sub * 16 + lrow;   // column of this lane
            const float cn = ntile[sub * 16 + lrow];
            #pragma unroll
            for (int g = 0; g < 8; ++g) {
                const float s = cn - 2.0f * acc[g];  // ||e||^2 - 2 z.e
                if (s < best[g]) { best[g] = s; bidx[g] = n; }
            }
        }
    }

    // Reduce argmin across the 16 column-lanes of each half-wave.
    #pragma unroll
    for (int g = 0; g < 8; ++g) {
        float s = best[g];
        int   i = bidx[g];
        #pragma unroll
        for (int m = 1; m < 16; m <<= 1) {
            const float s2 = __shfl_xor(s, m, 32);
            const int   i2 = __shfl_xor(i, m, 32);
            if (s2 < s || (s2 == s && i2 < i)) { s = s2; i = i2; }
        }
        if (lrow == 0) {
            const int r = g + half * 8;              // D-layout row
            out_idx[rowbase + r] = i;
        }
    }
}

// ---------------------------------------------------------------------------
// Kernel 3a: zero the loss accumulator
// ---------------------------------------------------------------------------
__global__ void vq_zero_loss_kernel(float* loss) { *loss = 0.f; }

// ---------------------------------------------------------------------------
// Kernel 3b: gather zq, permute to (B,C,H,W), accumulate loss
// loss = 1.25 * mean((zq - z)^2) over all B*C*H*W elements
// ---------------------------------------------------------------------------
__global__ __launch_bounds__(BLOCK)
void vq_out_kernel(const float* __restrict__ z,
                   const float* __restrict__ cb,
                   const int* __restrict__ idx,
                   float* __restrict__ out,
                   float* __restrict__ loss) {
    __shared__ float red[BLOCK];
    const int o = blockIdx.x * BLOCK + threadIdx.x;     // output linear index
    const int ww = o & 31, hh = (o >> 5) & 31, cc = (o >> 10) & 255, bb = o >> 18;
    const int pp = cc >> 6, dd = cc & 63;
    const int j = ((bb * P_ + pp) * H_ + hh) * W_ + ww; // zf row
    const float zq = cb[(size_t)idx[j] * ED + dd];
    const float zv = z[o];
    out[o] = zq;
    const float diff = zq - zv;
    red[threadIdx.x] = diff * diff;
    __syncthreads();
    #pragma unroll
    for (int s = BLOCK / 2; s > 0; s >>= 1) {
        if (threadIdx.x < s) red[threadIdx.x] += red[threadIdx.x + s];
        __syncthreads();
    }
    if (threadIdx.x == 0)
        atomicAdd(loss, red[0] * (1.25f / (float)NOUT));
}

// ---------------------------------------------------------------------------
extern "C" void kernel_launch(void* const* d_in, const int* in_sizes, int n_in,
                              void* d_out, int out_size, void* d_ws, size_t ws_size,
                              hipStream_t stream) {
    const float* z  = (const float*)d_in[0];   // [8,256,32,32] f32
    const float* cb = (const float*)d_in[1];   // [8192,64]     f32
    float* out  = (float*)d_out;               // 2097152 zq values + 1 loss
    float* loss = out + (out_size - 1);

    float* norms = (float*)d_ws;                             // 8192 f32
    int*   idx   = (int*)((char*)d_ws + NE * sizeof(float)); // 32768 i32

    vq_norms_kernel<<<NE / BLOCK, BLOCK, 0, stream>>>(cb, norms);
    vq_argmin_kernel<<<NROW / MTILE, BLOCK, 0, stream>>>(z, cb, norms, idx);
    vq_zero_loss_kernel<<<1, 1, 0, stream>>>(loss);
    vq_out_kernel<<<NOUT / BLOCK, BLOCK, 0, stream>>>(z, cb, idx, out, loss);
}